// SEDR_module_5428838662295
// MI455X (gfx1250) — compile-verified
//
#include <hip/hip_runtime.h>
#include <math.h>

#define NN    40000
#define DD    2500
#define IMGD  1024
#define EE    320000
#define FH1   64
#define FH2   16
#define GH1   64
#define GH2   16
#define CLW   32
#define KCL   10
#define ZW    32          // FH2 + GH2
#define MASKN 32000

typedef float v2f __attribute__((ext_vector_type(2)));
typedef float v8f __attribute__((ext_vector_type(8)));

// ---------------------------------------------------------------------------
// CDNA5 async global->LDS copy (ASYNCcnt) + drain. lds_off is the raw LDS
// byte offset (low 32 bits of the flat shared-aperture address).
// ---------------------------------------------------------------------------
__device__ __forceinline__ void async_g2l_b32(unsigned lds_off, const float* g)
{
    asm volatile("global_load_async_to_lds_b32 %0, %1, off"
                 :: "v"(lds_off), "v"((unsigned long long)(uintptr_t)g)
                 : "memory");
}
__device__ __forceinline__ void wait_async0()
{
    asm volatile("s_wait_asynccnt 0x0" ::: "memory");
}

// ---------------------------------------------------------------------------
// Skinny GEMM: C[M x Ncols] = op(A[M x K]) @ B[K x Ncols] (+ bias)
// Block = 8 waves: one 16-col N-tile shared, 8 consecutive 16-row M-tiles.
// B staged per 32-K chunk into LDS (col-major, pad 34) via async copies;
// each wave runs V_WMMA_F32_16X16X4_F32 over the chunk with ds_load_b64 B
// fragments and global_load_b64 A fragments.
// op(A): compile-time mask-token add (A + cnt[row]*mt[k]) and/or relu.
// M multiple of 16, K multiple of 4 at every call site; Ncols clamped.
// ---------------------------------------------------------------------------
template<bool MASKADD, bool RELUA>
__global__ __launch_bounds__(256) void gemm16_wmma(
    const float* __restrict__ A, const float* __restrict__ B,
    const float* __restrict__ bias, float* __restrict__ C,
    int M, int K, int Ncols, int tilesN, int tilesM,
    const float* __restrict__ cnt, const float* __restrict__ mt)
{
    __shared__ float bs[16 * 34];            // 16 cols x 32 k, padded stride 34

    int bN  = blockIdx.x % tilesN;
    int bM  = blockIdx.x / tilesN;
    int wid = threadIdx.x >> 5;
    int tM  = bM * 8 + wid;
    bool mvalid = tM < tilesM;
    int tMc = mvalid ? tM : tilesM - 1;      // clamp; idle waves still barrier

    int lane = threadIdx.x & 31;
    int half = lane >> 4;                    // 0: K={0,1}, 1: K={2,3}
    int l15  = lane & 15;
    int mrow = tMc * 16 + l15;
    int ncol = bN * 16 + l15;
    bool cok = ncol < Ncols;

    const float* arow = A + (size_t)mrow * K;
    float cscale = MASKADD ? cnt[mrow] : 0.0f;

    // B staging: thread t fills elements (fcol, fk) and (fcol, fk+1)
    int fcol = threadIdx.x & 15;
    int fk   = (threadIdx.x >> 4) << 1;      // 0,2,...,30
    int gcol = min(bN * 16 + fcol, Ncols - 1);

    v8f acc = {};
    for (int kb = 0; kb < K; kb += 32) {
        int chunk = min(32, K - kb);
        __syncthreads();                      // previous chunk fully consumed
        if (fk < chunk) {
            unsigned l0 = (unsigned)(uintptr_t)&bs[fcol * 34 + fk];
            async_g2l_b32(l0,     B + (size_t)(kb + fk)     * Ncols + gcol);
            async_g2l_b32(l0 + 4, B + (size_t)(kb + fk + 1) * Ncols + gcol);
        }
        wait_async0();
        __syncthreads();                      // all waves' copies visible
        #pragma unroll 4
        for (int ks = 0; ks < chunk; ks += 4) {
            int k0 = kb + ks + half * 2;
            v2f av = *(const v2f*)&arow[k0];              // global_load_b64
            if (MASKADD) {
                v2f mv = *(const v2f*)&mt[k0];
                av.x = fmaf(cscale, mv.x, av.x);
                av.y = fmaf(cscale, mv.y, av.y);
            }
            if (RELUA) {
                av.x = fmaxf(av.x, 0.0f);
                av.y = fmaxf(av.y, 0.0f);
            }
            v2f bv = *(const v2f*)&bs[l15 * 34 + ks + half * 2]; // ds_load_b64
            acc = __builtin_amdgcn_wmma_f32_16x16x4_f32(
                false, av, false, bv, (short)0, acc, false, false);
        }
    }

    if (mvalid && cok) {
        float bb = bias ? bias[ncol] : 0.0f;
        int rbase = tMc * 16 + half * 8;     // C layout: VGPR v -> row rbase+v
        float* cc = C + (size_t)rbase * Ncols + ncol;
        #pragma unroll
        for (int v = 0; v < 8; ++v)
            cc[(size_t)v * Ncols] = acc[v] + bb;
    }
}

// ---------------------------------------------------------------------------
// Per-node mask-token add count (both mask passes; duplicates accumulate)
// ---------------------------------------------------------------------------
__global__ void mask_count(const int* __restrict__ m1, const int* __restrict__ m2,
                           float* __restrict__ cnt)
{
    int i = blockIdx.x * blockDim.x + threadIdx.x;
    if (i < MASKN)          atomicAdd(&cnt[m1[i]], 1.0f);
    else if (i < 2 * MASKN) atomicAdd(&cnt[m2[i - MASKN]], 1.0f);
}

// ---------------------------------------------------------------------------
// BatchNorm column statistics: sums[c] += sum(H[:,c]), sums[C+c] += sum(H^2)
// ---------------------------------------------------------------------------
__global__ __launch_bounds__(256) void bn_stats(
    const float* __restrict__ H, float* __restrict__ sums, int C, int rowsPerBlock)
{
    int tpc = blockDim.x / C;
    int c   = threadIdx.x % C;
    int sub = threadIdx.x / C;
    int r0  = blockIdx.x * rowsPerBlock;
    float s = 0.0f, s2 = 0.0f;
    for (int r = r0 + sub; r < r0 + rowsPerBlock && r < NN; r += tpc) {
        float h = H[(size_t)r * C + c];
        s += h; s2 += h * h;
    }
    __shared__ float ls[256], lq[256];
    ls[threadIdx.x] = s; lq[threadIdx.x] = s2;
    __syncthreads();
    if (sub == 0) {
        for (int k = 1; k < tpc; ++k) { s += ls[k * C + c]; s2 += lq[k * C + c]; }
        atomicAdd(&sums[c], s);
        atomicAdd(&sums[C + c], s2);
    }
}

// BN normalize (biased var, eps=1e-3) + gamma/beta + ELU(alpha=1)
__global__ void bn_elu(const float* __restrict__ H, float* __restrict__ Dst,
                       const float* __restrict__ sums, const float* __restrict__ g,
                       const float* __restrict__ beta, int C)
{
    size_t i = (size_t)blockIdx.x * blockDim.x + threadIdx.x;
    if (i >= (size_t)NN * C) return;
    int c = (int)(i % C);
    float mu  = sums[c] * (1.0f / NN);
    float var = sums[C + c] * (1.0f / NN) - mu * mu;
    float h = (H[i] - mu) * rsqrtf(var + 1e-3f) * g[c] + beta[c];
    Dst[i] = (h > 0.0f) ? h : (__expf(h) - 1.0f);
}

// ---------------------------------------------------------------------------
// SpMM via edge-parallel atomics: one thread per (edge, feature) element.
// Out must be pre-zeroed.
// ---------------------------------------------------------------------------
template<int W>
__global__ __launch_bounds__(256) void spmm_edges(
    const int* __restrict__ idx, const float* __restrict__ vals,
    const float* __restrict__ S, float* __restrict__ Out)
{
    int gt = blockIdx.x * blockDim.x + threadIdx.x;
    int e  = gt / W;
    int j  = gt % W;
    if (e >= EE) return;
    int r = idx[e];
    int c = idx[EE + e];
    atomicAdd(&Out[(size_t)r * W + j], vals[e] * S[(size_t)c * W + j]);
}

// ---------------------------------------------------------------------------
// Reparameterize + attention fusion + build z = [feat_x | gnn_z]
// ---------------------------------------------------------------------------
__global__ void attn_fuse(const float* __restrict__ mu1, const float* __restrict__ lv1,
                          const float* __restrict__ mu2, const float* __restrict__ lv2,
                          const float* __restrict__ eps1, const float* __restrict__ eps2,
                          const float* __restrict__ Wo, const float* __restrict__ Uo,
                          const float* __restrict__ featx,
                          float* __restrict__ z_out, float* __restrict__ gnnz_out)
{
    int n = blockIdx.x * blockDim.x + threadIdx.x;
    if (n >= NN) return;
    float z1[GH2], z2[GH2];
    #pragma unroll
    for (int h = 0; h < GH2; ++h) {
        int i = n * GH2 + h;
        z1[h] = eps1[i] * __expf(lv1[i]) + mu1[i];
        z2[h] = eps2[i] * __expf(lv2[i]) + mu2[i];
    }
    float vu1 = 0.0f, vu2 = 0.0f;
    for (int j = 0; j < GH2; ++j) {
        float a1 = 0.0f, a2 = 0.0f;
        #pragma unroll
        for (int h = 0; h < GH2; ++h) {
            float w = Wo[h * GH2 + j];
            a1 = fmaf(z1[h], w, a1);
            a2 = fmaf(z2[h], w, a2);
        }
        vu1 = fmaf(tanhf(a1), Uo[j], vu1);
        vu2 = fmaf(tanhf(a2), Uo[j], vu2);
    }
    vu1 += 1e-6f; vu2 += 1e-6f;
    float m = fmaxf(vu1, vu2);
    float e1 = __expf(vu1 - m), e2 = __expf(vu2 - m);
    float inv = 1.0f / (e1 + e2);
    float w1 = e1 * inv, w2 = e2 * inv;
    #pragma unroll
    for (int h = 0; h < GH2; ++h) {
        float gz = w1 * z1[h] + w2 * z2[h];
        gnnz_out[n * GH2 + h] = gz;
        z_out[(size_t)n * ZW + h]       = featx[n * FH2 + h];
        z_out[(size_t)n * ZW + FH2 + h] = gz;
    }
}

// DEC soft assignment q (alpha = 1 => power exponent is identity)
__global__ void dec_q(const float* __restrict__ z, const float* __restrict__ cluster,
                      float* __restrict__ q)
{
    int n = blockIdx.x * blockDim.x + threadIdx.x;
    if (n >= NN) return;
    float zr[ZW];
    #pragma unroll
    for (int h = 0; h < ZW; ++h) zr[h] = z[(size_t)n * ZW + h];
    float qs[KCL]; float tot = 0.0f;
    for (int k = 0; k < KCL; ++k) {
        float sq = 0.0f;
        #pragma unroll
        for (int h = 0; h < ZW; ++h) { float d = zr[h] - cluster[k * ZW + h]; sq = fmaf(d, d, sq); }
        float qq = 1.0f / (1.0f + sq);
        qs[k] = qq; tot += qq;
    }
    float inv = 1.0f / tot;
    for (int k = 0; k < KCL; ++k) q[(size_t)n * KCL + k] = qs[k] * inv;
}

// ---------------------------------------------------------------------------
// SCE loss: block per masked row, float4 (B128) streaming, 5-way dot reduce.
// DD = 2500 = 625 float4; every row base is 16B aligned (2500*4 % 16 == 0).
// ---------------------------------------------------------------------------
__global__ __launch_bounds__(256) void sce_loss(
    const int* __restrict__ m2, const float* __restrict__ df1,
    const float* __restrict__ df2, const float* __restrict__ x,
    const float* __restrict__ cnt, const float* __restrict__ mt,
    float* __restrict__ acc)
{
    int row = m2[blockIdx.x];
    const float4* a1 = (const float4*)(df1 + (size_t)row * DD);
    const float4* a2 = (const float4*)(df2 + (size_t)row * DD);
    const float4* xr = (const float4*)(x   + (size_t)row * DD);
    const float4* m4 = (const float4*)mt;
    float c = cnt[row];
    float xx = 0, d1x = 0, d11 = 0, d2x = 0, d22 = 0;
    for (int j = threadIdx.x; j < DD / 4; j += blockDim.x) {
        float4 xv = xr[j], mv = m4[j], v1 = a1[j], v2 = a2[j];
        xv.x = fmaf(c, mv.x, xv.x); xv.y = fmaf(c, mv.y, xv.y);
        xv.z = fmaf(c, mv.z, xv.z); xv.w = fmaf(c, mv.w, xv.w);
        xx  = fmaf(xv.x, xv.x, fmaf(xv.y, xv.y, fmaf(xv.z, xv.z, fmaf(xv.w, xv.w, xx))));
        d1x = fmaf(v1.x, xv.x, fmaf(v1.y, xv.y, fmaf(v1.z, xv.z, fmaf(v1.w, xv.w, d1x))));
        d11 = fmaf(v1.x, v1.x, fmaf(v1.y, v1.y, fmaf(v1.z, v1.z, fmaf(v1.w, v1.w, d11))));
        d2x = fmaf(v2.x, xv.x, fmaf(v2.y, xv.y, fmaf(v2.z, xv.z, fmaf(v2.w, xv.w, d2x))));
        d22 = fmaf(v2.x, v2.x, fmaf(v2.y, v2.y, fmaf(v2.z, v2.z, fmaf(v2.w, v2.w, d22))));
    }
    for (int off = 16; off > 0; off >>= 1) {
        xx  += __shfl_down(xx,  off, 32);
        d1x += __shfl_down(d1x, off, 32);
        d11 += __shfl_down(d11, off, 32);
        d2x += __shfl_down(d2x, off, 32);
        d22 += __shfl_down(d22, off, 32);
    }
    __shared__ float red[5][8];
    int lane = threadIdx.x & 31, w = threadIdx.x >> 5;
    if (lane == 0) { red[0][w]=xx; red[1][w]=d1x; red[2][w]=d11; red[3][w]=d2x; red[4][w]=d22; }
    __syncthreads();
    if (threadIdx.x == 0) {
        int nw = blockDim.x >> 5;
        xx=0; d1x=0; d11=0; d2x=0; d22=0;
        for (int k = 0; k < nw; ++k) {
            xx+=red[0][k]; d1x+=red[1][k]; d11+=red[2][k]; d2x+=red[3][k]; d22+=red[4][k];
        }
        float nx = fmaxf(sqrtf(xx),  1e-12f);
        float n1 = fmaxf(sqrtf(d11), 1e-12f);
        float n2 = fmaxf(sqrtf(d22), 1e-12f);
        float c1 = 1.0f - d1x / (n1 * nx);
        float c2 = 1.0f - d2x / (n2 * nx);
        atomicAdd(acc, c1 * c1 * c1 + c2 * c2 * c2);
    }
}

__global__ void loss_finalize(const float* __restrict__ acc, float* __restrict__ out)
{
    if (blockIdx.x == 0 && threadIdx.x == 0) out[0] = acc[0] * (1.0f / MASKN);
}

// ---------------------------------------------------------------------------
// Host launcher
// ---------------------------------------------------------------------------
extern "C" void kernel_launch(void* const* d_in, const int* in_sizes, int n_in,
                              void* d_out, int out_size, void* d_ws, size_t ws_size,
                              hipStream_t stream)
{
    (void)in_sizes; (void)n_in; (void)out_size; (void)ws_size;

    const float* x        = (const float*)d_in[0];
    const float* imgf     = (const float*)d_in[1];
    const int*   adj1_idx = (const int*)  d_in[2];
    const float* adj1_val = (const float*)d_in[3];
    const int*   adj2_idx = (const int*)  d_in[4];
    const float* adj2_val = (const float*)d_in[5];
    const int*   mask1    = (const int*)  d_in[6];
    const int*   mask2    = (const int*)  d_in[7];
    const float* eps1     = (const float*)d_in[8];
    const float* eps2     = (const float*)d_in[9];
    const float* enc1_W   = (const float*)d_in[10];
    const float* enc1_b   = (const float*)d_in[11];
    const float* enc1_g   = (const float*)d_in[12];
    const float* enc1_be  = (const float*)d_in[13];
    const float* enc2_W   = (const float*)d_in[14];
    const float* enc2_b   = (const float*)d_in[15];
    const float* enc2_g   = (const float*)d_in[16];
    const float* enc2_be  = (const float*)d_in[17];
    const float* gc1_W    = (const float*)d_in[18];
    const float* gc2_W    = (const float*)d_in[19];
    const float* gc3_W    = (const float*)d_in[20];
    const float* dec_W    = (const float*)d_in[21];
    const float* w_omega  = (const float*)d_in[22];
    const float* u_omega  = (const float*)d_in[23];
    const float* fcr_W    = (const float*)d_in[24];
    const float* fcr_b    = (const float*)d_in[25];
    const float* fci_W    = (const float*)d_in[26];
    const float* fci_b    = (const float*)d_in[27];
    const float* cluster  = (const float*)d_in[28];
    const float* mtok     = (const float*)d_in[29];

    float* out = (float*)d_out;
    const size_t OFF_Z   = 0;
    const size_t OFF_MU1 = OFF_Z   + (size_t)NN * ZW;
    const size_t OFF_LV1 = OFF_MU1 + (size_t)NN * GH2;
    const size_t OFF_MU2 = OFF_LV1 + (size_t)NN * GH2;
    const size_t OFF_LV2 = OFF_MU2 + (size_t)NN * GH2;
    const size_t OFF_DF1 = OFF_LV2 + (size_t)NN * GH2;
    const size_t OFF_DF2 = OFF_DF1 + (size_t)NN * DD;
    const size_t OFF_Q   = OFF_DF2 + (size_t)NN * DD;
    const size_t OFF_FX  = OFF_Q   + (size_t)NN * KCL;
    const size_t OFF_GZ  = OFF_FX  + (size_t)NN * FH2;
    const size_t OFF_L   = OFF_GZ  + (size_t)NN * GH2;
    const size_t OFF_RNA = OFF_L   + 1;
    const size_t OFF_IMG = OFF_RNA + (size_t)NN * ZW;

    float* ws = (float*)d_ws;
    size_t o = 0;
    float* cnt   = ws + o; o += NN;
    float* hpre1 = ws + o; o += (size_t)NN * FH1;
    float* tbuf  = ws + o; o += (size_t)NN * GH1;
    float* h1    = ws + o; o += (size_t)NN * GH1;
    float* h2    = ws + o; o += (size_t)NN * GH1;
    float* hpre2 = ws + o; o += (size_t)NN * FH2;
    float* ubuf  = ws + o; o += (size_t)NN * GH2;
    float* agg   = ws + o; o += (size_t)NN * ZW;
    float* stats = ws + o; o += 256;
    float* lacc  = ws + o; o += 16;

    auto gemm = [&](const float* A, const float* B, const float* bias, float* C,
                    int M, int K, int Ncols, const float* cp, const float* mp,
                    bool maskadd, bool relu) {
        int tilesN = (Ncols + 15) / 16;
        int tilesM = (M + 15) / 16;
        int blocks = tilesN * ((tilesM + 7) / 8);
        if (maskadd)
            gemm16_wmma<true, false><<<blocks, 256, 0, stream>>>(
                A, B, bias, C, M, K, Ncols, tilesN, tilesM, cp, mp);
        else if (relu)
            gemm16_wmma<false, true><<<blocks, 256, 0, stream>>>(
                A, B, bias, C, M, K, Ncols, tilesN, tilesM, nullptr, nullptr);
        else
            gemm16_wmma<false, false><<<blocks, 256, 0, stream>>>(
                A, B, bias, C, M, K, Ncols, tilesN, tilesM, nullptr, nullptr);
    };

    // ---- zero accumulators (deterministic across graph replays) ----
    hipMemsetAsync(cnt,  0, sizeof(float) * NN, stream);
    hipMemsetAsync(h1,   0, sizeof(float) * (size_t)NN * GH1, stream);
    hipMemsetAsync(h2,   0, sizeof(float) * (size_t)NN * GH1, stream);
    hipMemsetAsync(stats,0, sizeof(float) * 256, stream);
    hipMemsetAsync(lacc, 0, sizeof(float) * 16, stream);
    hipMemsetAsync(out + OFF_MU1, 0, sizeof(float) * (size_t)NN * GH2 * 4, stream);

    mask_count<<<(2 * MASKN + 255) / 256, 256, 0, stream>>>(mask1, mask2, cnt);

    // ---- encoder block 1: xm @ enc1_W + b -> BN -> ELU ----
    gemm(x, enc1_W, enc1_b, hpre1, NN, DD, FH1, cnt, mtok, true, false);
    bn_stats<<<(NN + 1023) / 1024, 256, 0, stream>>>(hpre1, stats, FH1, 1024);
    bn_elu<<<(int)(((size_t)NN * FH1 + 255) / 256), 256, 0, stream>>>(
        hpre1, hpre1, stats, enc1_g, enc1_be, FH1);

    // ---- encoder block 2 -> feat_x (into its output slot) ----
    hipMemsetAsync(stats, 0, sizeof(float) * 256, stream);
    gemm(hpre1, enc2_W, enc2_b, hpre2, NN, FH1, FH2, nullptr, nullptr, false, false);
    bn_stats<<<(NN + 1023) / 1024, 256, 0, stream>>>(hpre2, stats, FH2, 1024);
    bn_elu<<<(int)(((size_t)NN * FH2 + 255) / 256), 256, 0, stream>>>(
        hpre2, out + OFF_FX, stats, enc2_g, enc2_be, FH2);

    // ---- GCN branches (t = feat_x @ gc1_W shared by both adjacencies) ----
    gemm(out + OFF_FX, gc1_W, nullptr, tbuf, NN, FH2, GH1, nullptr, nullptr, false, false);
    spmm_edges<GH1><<<(EE * GH1 + 255) / 256, 256, 0, stream>>>(adj1_idx, adj1_val, tbuf, h1);
    spmm_edges<GH1><<<(EE * GH1 + 255) / 256, 256, 0, stream>>>(adj2_idx, adj2_val, tbuf, h2);
    // relu folded into the A-operand of the following GEMMs
    gemm(h1, gc2_W, nullptr, ubuf, NN, GH1, GH2, nullptr, nullptr, false, true);
    spmm_edges<GH2><<<(EE * GH2 + 255) / 256, 256, 0, stream>>>(adj1_idx, adj1_val, ubuf, out + OFF_MU1);
    gemm(h1, gc3_W, nullptr, ubuf, NN, GH1, GH2, nullptr, nullptr, false, true);
    spmm_edges<GH2><<<(EE * GH2 + 255) / 256, 256, 0, stream>>>(adj1_idx, adj1_val, ubuf, out + OFF_LV1);
    gemm(h2, gc2_W, nullptr, ubuf, NN, GH1, GH2, nullptr, nullptr, false, true);
    spmm_edges<GH2><<<(EE * GH2 + 255) / 256, 256, 0, stream>>>(adj2_idx, adj2_val, ubuf, out + OFF_MU2);
    gemm(h2, gc3_W, nullptr, ubuf, NN, GH1, GH2, nullptr, nullptr, false, true);
    spmm_edges<GH2><<<(EE * GH2 + 255) / 256, 256, 0, stream>>>(adj2_idx, adj2_val, ubuf, out + OFF_LV2);

    // ---- reparameterize + attention fusion + z ----
    attn_fuse<<<(NN + 255) / 256, 256, 0, stream>>>(
        out + OFF_MU1, out + OFF_LV1, out + OFF_MU2, out + OFF_LV2,
        eps1, eps2, w_omega, u_omega, out + OFF_FX, out + OFF_Z, out + OFF_GZ);

    // ---- rna_emb = z @ fc_rna_W + b ----
    gemm(out + OFF_Z, fcr_W, fcr_b, out + OFF_RNA, NN, ZW, CLW, nullptr, nullptr, false, false);

    // ---- decoder: de_feat = (A @ rna_emb) @ dec_W  (linearity of segment_sum) ----
    hipMemsetAsync(agg, 0, sizeof(float) * (size_t)NN * ZW, stream);
    spmm_edges<ZW><<<(EE * ZW + 255) / 256, 256, 0, stream>>>(adj1_idx, adj1_val, out + OFF_RNA, agg);
    gemm(agg, dec_W, nullptr, out + OFF_DF1, NN, ZW, DD, nullptr, nullptr, false, false);
    hipMemsetAsync(agg, 0, sizeof(float) * (size_t)NN * ZW, stream);
    spmm_edges<ZW><<<(EE * ZW + 255) / 256, 256, 0, stream>>>(adj2_idx, adj2_val, out + OFF_RNA, agg);
    gemm(agg, dec_W, nullptr, out + OFF_DF2, NN, ZW, DD, nullptr, nullptr, false, false);

    // ---- image embedding ----
    gemm(imgf, fci_W, fci_b, out + OFF_IMG, NN, IMGD, CLW, nullptr, nullptr, false, false);

    // ---- DEC q ----
    dec_q<<<(NN + 255) / 256, 256, 0, stream>>>(out + OFF_Z, cluster, out + OFF_Q);

    // ---- SCE reconstruction loss on second mask ----
    sce_loss<<<MASKN, 256, 0, stream>>>(mask2, out + OFF_DF1, out + OFF_DF2,
                                        x, cnt, mtok, lacc);
    loss_finalize<<<1, 1, 0, stream>>>(lacc, out + OFF_L);
}